// SE_Non_local_81157702025718
// MI455X (gfx1250) — compile-verified
//
#include <hip/hip_runtime.h>
#include <math.h>

#define B_ 4
#define C_ 512
#define N_ 4096
#define R_ 32

typedef __attribute__((ext_vector_type(2))) float v2f;
typedef __attribute__((ext_vector_type(8))) float v8f;

// ---------------------------------------------------------------------------
// Kernel 1: per-pixel 1x1-conv projections g/theta/phi (channel dot products).
// Block = 256 threads = 256 pixels; grid = (N/256, B). Coalesced over pixels.
// ---------------------------------------------------------------------------
__global__ __launch_bounds__(256) void proj_kernel(
    const float* __restrict__ x,
    const float* __restrict__ g_w,  const float* __restrict__ g_b,
    const float* __restrict__ th_w, const float* __restrict__ th_b,
    const float* __restrict__ ph_w, const float* __restrict__ ph_b,
    float* __restrict__ theta_out, float* __restrict__ prod_out)
{
    __shared__ float wg[C_], wt[C_], wp[C_];
    const int t = threadIdx.x;
    for (int c = t; c < C_; c += 256) {
        wg[c] = g_w[c]; wt[c] = th_w[c]; wp[c] = ph_w[c];
    }
    __syncthreads();

    const int b = blockIdx.y;
    const int i = blockIdx.x * 256 + t;
    const float* px = x + ((size_t)b * C_) * N_ + i;

    float ag = 0.f, at = 0.f, ap = 0.f;
    #pragma unroll 8
    for (int c = 0; c < C_; ++c) {
        if (((c & 7) == 0) && (c + 32 < C_))
            __builtin_prefetch(px + (size_t)(c + 32) * N_, 0, 0);
        float v = px[(size_t)c * N_];
        ag = fmaf(v, wg[c], ag);
        at = fmaf(v, wt[c], at);
        ap = fmaf(v, wp[c], ap);
    }
    float gg = ag + g_b[0];
    float tt = at + th_b[0];
    float pp = ap + ph_b[0];
    theta_out[b * N_ + i] = tt;
    prod_out[b * N_ + i]  = gg * pp;   // elementwise product for rank-1 dot
}

// ---------------------------------------------------------------------------
// Kernel 2: scal[b] = dot(phi, g) / N  (fixed-order deterministic reduction)
// ---------------------------------------------------------------------------
__global__ __launch_bounds__(256) void scal_kernel(
    const float* __restrict__ prod, float* __restrict__ scal)
{
    __shared__ float red[256];
    const int b = blockIdx.x, t = threadIdx.x;
    float s = 0.f;
    #pragma unroll
    for (int k = 0; k < N_ / 256; ++k) s += prod[b * N_ + k * 256 + t];
    red[t] = s;
    __syncthreads();
    for (int w = 128; w > 0; w >>= 1) {
        if (t < w) red[t] += red[t + w];
        __syncthreads();
    }
    if (t == 0) scal[b] = red[0] * (1.0f / (float)N_);
}

// ---------------------------------------------------------------------------
// Kernel 3: s = x + theta*scal*(w_w*inv) + ((w_b-mean)*inv + beta)
//           also pooled[b,c] = mean(s) over spatial dims.
// One block per (b,c) row of 4096 contiguous floats; float4 vectorized.
// ---------------------------------------------------------------------------
__global__ __launch_bounds__(256) void fuse_kernel(
    const float* __restrict__ x, const float* __restrict__ theta,
    const float* __restrict__ scal,
    const float* __restrict__ w_w, const float* __restrict__ w_b,
    const float* __restrict__ bn_gamma, const float* __restrict__ bn_beta,
    const float* __restrict__ bn_mean,  const float* __restrict__ bn_var,
    float* __restrict__ s_out, float* __restrict__ pooled)
{
    __shared__ float red[256];
    const int bc = blockIdx.x;
    const int b  = bc >> 9;
    const int c  = bc & (C_ - 1);
    const int t  = threadIdx.x;

    const float inv = bn_gamma[c] / sqrtf(bn_var[c] + 1e-5f);
    const float a   = w_w[c] * inv;
    const float off = (w_b[c] - bn_mean[c]) * inv + bn_beta[c];
    const float ya  = scal[b] * a;

    const float4* x4 = (const float4*)(x     + (size_t)bc * N_);
    const float4* t4 = (const float4*)(theta + (size_t)b  * N_);
    float4*       o4 = (float4*)(s_out       + (size_t)bc * N_);

    float lsum = 0.f;
    #pragma unroll
    for (int k = 0; k < 4; ++k) {
        int j = k * 256 + t;
        float4 xv = x4[j], tv = t4[j], sv;
        sv.x = fmaf(tv.x, ya, off) + xv.x;
        sv.y = fmaf(tv.y, ya, off) + xv.y;
        sv.z = fmaf(tv.z, ya, off) + xv.z;
        sv.w = fmaf(tv.w, ya, off) + xv.w;
        o4[j] = sv;
        lsum += (sv.x + sv.y) + (sv.z + sv.w);
    }
    red[t] = lsum;
    __syncthreads();
    for (int w = 128; w > 0; w >>= 1) {
        if (t < w) red[t] += red[t + w];
        __syncthreads();
    }
    if (t == 0) pooled[bc] = red[0] * (1.0f / (float)N_);
}

// ---------------------------------------------------------------------------
// Kernel 4: SE MLP with V_WMMA_F32_16X16X4_F32, single wave32.
//   h    = relu(pooled[16pad,512] @ fc1_w[32,512]^T)      (128 k-steps x 2 tiles)
//   gate = sigmoid(h[16pad,32]   @ fc2_w[512,32]^T)       (32 tiles x 8 k-steps)
// A layout 16x4 f32: lanes 0-15 -> K={k0,k0+1}, lanes 16-31 -> K={k0+2,k0+3}.
// C/D layout: lane L, vgpr v -> M = v + 8*(L>=16), N = L&15.
// Padded batch rows 4..15: load row (m&3) unconditionally (always in-bounds)
// and multiply by a 0/1 lane mask -> no divergent predicated loads.
// ---------------------------------------------------------------------------
__global__ __launch_bounds__(32) void se_gate_kernel(
    const float* __restrict__ pooled,   // [4,512] valid rows 0..3
    const float* __restrict__ fc1_w,    // [32,512]
    const float* __restrict__ fc2_w,    // [512,32]
    float* __restrict__ gate)           // [4,512]
{
    __shared__ float h_lds[16 * R_];
    const int lane  = threadIdx.x;      // 0..31
    const int m     = lane & 15;        // row / col-in-tile index
    const int khalf = (lane >> 4) * 2;  // 0 or 2: K sub-pair per lane half
    const float rowmask = (m < 4) ? 1.0f : 0.0f;

    // ---- fc1: two 16-wide N tiles, K = 512 ----
    v8f acc0 = {}; v8f acc1 = {};
    const float* prow = pooled + (m & 3) * C_;   // always in-bounds
    for (int k0 = 0; k0 < C_; k0 += 4) {
        v2f a;
        a.x = prow[k0 + khalf]     * rowmask;
        a.y = prow[k0 + khalf + 1] * rowmask;
        v2f b0, b1;
        b0.x = fc1_w[m * C_        + k0 + khalf];
        b0.y = fc1_w[m * C_        + k0 + khalf + 1];
        b1.x = fc1_w[(m + 16) * C_ + k0 + khalf];
        b1.y = fc1_w[(m + 16) * C_ + k0 + khalf + 1];
        acc0 = __builtin_amdgcn_wmma_f32_16x16x4_f32(false, a, false, b0,
                                                     (short)0, acc0, false, false);
        acc1 = __builtin_amdgcn_wmma_f32_16x16x4_f32(false, a, false, b1,
                                                     (short)0, acc1, false, false);
    }

    // ReLU, convert D-layout -> row-major h in LDS (16x32)
    const int rbase = (lane >> 4) * 8;
    const int ncol  = lane & 15;
    #pragma unroll
    for (int v = 0; v < 8; ++v) {
        float h0 = acc0[v]; h0 = h0 > 0.f ? h0 : 0.f;
        float h1 = acc1[v]; h1 = h1 > 0.f ? h1 : 0.f;
        h_lds[(rbase + v) * R_ + ncol]      = h0;
        h_lds[(rbase + v) * R_ + 16 + ncol] = h1;
    }
    __syncthreads();

    // ---- fc2: 32 N-tiles of 16, K = 32 ----
    for (int tile = 0; tile < C_ / 16; ++tile) {
        v8f d = {};
        #pragma unroll
        for (int ks = 0; ks < R_ / 4; ++ks) {
            int kk = ks * 4 + khalf;
            v2f a, b;
            a.x = h_lds[m * R_ + kk];
            a.y = h_lds[m * R_ + kk + 1];
            int n = tile * 16 + m;
            b.x = fc2_w[n * R_ + kk];
            b.y = fc2_w[n * R_ + kk + 1];
            d = __builtin_amdgcn_wmma_f32_16x16x4_f32(false, a, false, b,
                                                      (short)0, d, false, false);
        }
        if (lane < 16) {               // rows 0..3 live in vgprs 0..3, lanes 0..15
            int n = tile * 16 + lane;
            #pragma unroll
            for (int v = 0; v < 4; ++v) {
                float z = d[v];
                gate[v * C_ + n] = 1.0f / (1.0f + expf(-z));
            }
        }
    }
}

// ---------------------------------------------------------------------------
// Kernel 5: out = s * gate[b,c] (in place over d_out)
// ---------------------------------------------------------------------------
__global__ __launch_bounds__(256) void scale_kernel(
    float* __restrict__ out, const float* __restrict__ gate)
{
    const int bc = blockIdx.x;
    const int t  = threadIdx.x;
    const float g = gate[bc];
    float4* o4 = (float4*)(out + (size_t)bc * N_);
    #pragma unroll
    for (int k = 0; k < 4; ++k) {
        int j = k * 256 + t;
        float4 v = o4[j];
        v.x *= g; v.y *= g; v.z *= g; v.w *= g;
        o4[j] = v;
    }
}

// ---------------------------------------------------------------------------
extern "C" void kernel_launch(void* const* d_in, const int* in_sizes, int n_in,
                              void* d_out, int out_size, void* d_ws, size_t ws_size,
                              hipStream_t stream)
{
    const float* x        = (const float*)d_in[0];
    const float* g_w      = (const float*)d_in[1];
    const float* g_b      = (const float*)d_in[2];
    const float* th_w     = (const float*)d_in[3];
    const float* th_b     = (const float*)d_in[4];
    const float* ph_w     = (const float*)d_in[5];
    const float* ph_b     = (const float*)d_in[6];
    const float* w_w      = (const float*)d_in[7];
    const float* w_b      = (const float*)d_in[8];
    const float* bn_gamma = (const float*)d_in[9];
    const float* bn_beta  = (const float*)d_in[10];
    const float* bn_mean  = (const float*)d_in[11];
    const float* bn_var   = (const float*)d_in[12];
    const float* fc1_w    = (const float*)d_in[13];
    const float* fc2_w    = (const float*)d_in[14];

    float* out = (float*)d_out;
    float* ws  = (float*)d_ws;
    // workspace layout (floats): ~148 KB total
    float* theta  = ws;              // B*N   = 16384
    float* prod   = ws + 16384;      // B*N   = 16384
    float* scal   = ws + 32768;      // B     = 4
    float* pooled = ws + 32800;      // B*C   = 2048
    float* gate   = ws + 34848;      // B*C   = 2048

    proj_kernel<<<dim3(N_ / 256, B_), 256, 0, stream>>>(
        x, g_w, g_b, th_w, th_b, ph_w, ph_b, theta, prod);
    scal_kernel<<<B_, 256, 0, stream>>>(prod, scal);
    fuse_kernel<<<B_ * C_, 256, 0, stream>>>(
        x, theta, scal, w_w, w_b, bn_gamma, bn_beta, bn_mean, bn_var, out, pooled);
    se_gate_kernel<<<1, 32, 0, stream>>>(pooled, fc1_w, fc2_w, gate);
    scale_kernel<<<B_ * C_, 256, 0, stream>>>(out, gate);
}